// HierarchyStructureLoss_46411416601015
// MI455X (gfx1250) — compile-verified
//
#include <hip/hip_runtime.h>

typedef __attribute__((ext_vector_type(2))) float v2f;
typedef __attribute__((ext_vector_type(8))) float v8f;
typedef __attribute__((ext_vector_type(4))) int   v4i;
typedef __attribute__((address_space(1))) v4i     g_v4i;   // global int4
typedef __attribute__((address_space(3))) v4i     s_v4i;   // LDS int4

#define NUM_CLASSES 50000
#define CDIM 512
#define N_PC 200000
#define N_SIB 200000
#define PARENT_MARGIN 0.3f
#define SIBLING_MARGIN 0.5f

#define KCHUNK 32
#define ROWPAD (KCHUNK + 4)       // 36 floats/row: 16B-aligned rows, conflict-free wmma reads
#define WAVES_PER_BLOCK 4
#define TILE 16                   // pairs per wave-tile
#define NT_PC (N_PC / TILE)       // 12500
#define NT_TOTAL ((N_PC + N_SIB) / TILE)  // 25000
#define NSLOT 256
#define SLOT_STRIDE 32            // floats; one 128B line per accumulator slot
#define ASYNC_PER_STAGE 8         // async-to-LDS instructions issued per chunk stage

#if defined(__has_builtin)
#  if __has_builtin(__builtin_amdgcn_global_load_async_to_lds_b128)
#    define HAVE_ASYNC_LDS 1
#  else
#    define HAVE_ASYNC_LDS 0
#  endif
#else
#  define HAVE_ASYNC_LDS 0
#endif

template <int N>
__device__ __forceinline__ void wait_async() {
#if defined(__has_builtin) && __has_builtin(__builtin_amdgcn_s_wait_asynccnt)
    __builtin_amdgcn_s_wait_asynccnt(N);
#else
    asm volatile("s_wait_asynccnt %0" :: "i"(N) : "memory");
#endif
}

__device__ __forceinline__ void wait_ds0() {
#if defined(__has_builtin) && __has_builtin(__builtin_amdgcn_s_wait_dscnt)
    __builtin_amdgcn_s_wait_dscnt(0);
#else
    asm volatile("s_wait_dscnt 0x0" ::: "memory");
#endif
}

__global__ void zero_kernel(float* slots) {
    const int i = blockIdx.x * blockDim.x + threadIdx.x;
    if (i < NSLOT * SLOT_STRIDE) slots[i] = 0.0f;
}

// One wave per embedding row: sum of squares, shfl-xor reduce, inv_norm = 1/max(||x||, eps)
__global__ void inv_norm_kernel(const float* __restrict__ emb,
                                float* __restrict__ inv_norm) {
    const int row  = (int)((blockIdx.x * blockDim.x + threadIdx.x) >> 5);
    const int lane = threadIdx.x & 31;
    if (row >= NUM_CLASSES) return;
    const float4* r4 = (const float4*)(emb + (size_t)row * CDIM);
    float ss = 0.0f;
#pragma unroll
    for (int i = 0; i < 4; ++i) {
        float4 v = r4[lane + 32 * i];
        ss += v.x * v.x + v.y * v.y + v.z * v.z + v.w * v.w;
    }
#pragma unroll
    for (int off = 16; off > 0; off >>= 1) ss += __shfl_xor(ss, off, 32);
    if (lane == 0) inv_norm[row] = 1.0f / fmaxf(sqrtf(ss), 1e-12f);
}

// Stage one K-chunk (16 a-rows + 16 b-rows x KCHUNK floats) into per-wave LDS from
// precomputed per-lane row base pointers. 8 async-to-LDS ops, no index math in loop.
__device__ __forceinline__ void stage_chunk(const float* const* ga, const float* const* gb,
                                            int lane, int kc,
                                            float* la_base, float* lb_base) {
    const int c4 = lane & 7;          // float4 column within chunk
#pragma unroll
    for (int i = 0; i < 4; ++i) {
        const int row = 4 * i + (lane >> 3);     // 0..15
        float* la = la_base + row * ROWPAD + 4 * c4;
        float* lb = lb_base + row * ROWPAD + 4 * c4;
#if HAVE_ASYNC_LDS
        __builtin_amdgcn_global_load_async_to_lds_b128((g_v4i*)(ga[i] + kc), (s_v4i*)la, 0, 0);
        __builtin_amdgcn_global_load_async_to_lds_b128((g_v4i*)(gb[i] + kc), (s_v4i*)lb, 0, 0);
#else
        *(float4*)la = *(const float4*)(ga[i] + kc);
        *(float4*)lb = *(const float4*)(gb[i] + kc);
#endif
    }
}

// 8 x V_WMMA_F32_16X16X4_F32 over one staged chunk (exact fp32).
__device__ __forceinline__ void consume_chunk(const float* pa, const float* pb, v8f& acc) {
#pragma unroll
    for (int t = 0; t < KCHUNK / 4; ++t) {
        v2f a = *(const v2f*)(pa + 4 * t);
        v2f b = *(const v2f*)(pb + 4 * t);
        acc = __builtin_amdgcn_wmma_f32_16x16x4_f32(
            false, a, false, b, (short)0, acc, false, false);
    }
}

// One wave per 16-pair tile, barrier-free, double-buffered async pipeline:
// stage chunk k+1 (ASYNCcnt) while running WMMA on chunk k; last chunk peeled so the
// steady-state loop body is branch-free.
__global__ __launch_bounds__(WAVES_PER_BLOCK * 32)
void pair_loss_kernel(const float* __restrict__ emb,
                      const int* __restrict__ pc,
                      const int* __restrict__ sib,
                      const float* __restrict__ inv_norm,
                      float* __restrict__ slots) {
    __shared__ float ldsA[2][WAVES_PER_BLOCK][16 * ROWPAD];
    __shared__ float ldsB[2][WAVES_PER_BLOCK][16 * ROWPAD];

    const int wave = threadIdx.x >> 5;
    const int lane = threadIdx.x & 31;
    const int tile = blockIdx.x * WAVES_PER_BLOCK + wave;   // grid sized exactly

    const bool  is_pc  = tile < NT_PC;
    const int*  pairs  = is_pc ? pc : sib;
    const int   tbase  = is_pc ? tile : (tile - NT_PC);
    const float margin = is_pc ? PARENT_MARGIN : SIBLING_MARGIN;

    // lanes 0..15 own column 0 (a) of pair (lane&15); lanes 16..31 own column 1 (b)
    const int   idxOwn = pairs[2 * (tbase * TILE + (lane & 15)) + (lane >> 4)];
    const float invOwn = inv_norm[idxOwn];

    // Hoist gather addressing out of the K loop: per-lane quad base pointers for the
    // 4 a-rows / 4 b-rows this lane stages (shfl broadcasts run exactly once).
    const float* ga[4];
    const float* gb[4];
    {
        const int c4 = lane & 7;
#pragma unroll
        for (int i = 0; i < 4; ++i) {
            const int row = 4 * i + (lane >> 3);
            const int ia  = __shfl(idxOwn, row, 32);       // a-index lives at lane row
            const int ib  = __shfl(idxOwn, 16 + row, 32);  // b-index lives at lane 16+row
            ga[i] = emb + (size_t)ia * CDIM + 4 * c4;
            gb[i] = emb + (size_t)ib * CDIM + 4 * c4;
        }
    }

    v8f acc = {};
    const int r  = lane & 15;             // row within tile (M for A-frag, N for B-frag)
    const int hi = (lane >> 4) << 1;      // K sub-offset: 0 (lanes 0-15) or 2 (lanes 16-31)

    stage_chunk(ga, gb, lane, 0, &ldsA[0][wave][0], &ldsB[0][wave][0]);

    int cur = 0;
#pragma unroll 1
    for (int kc = 0; kc < CDIM - KCHUNK; kc += KCHUNK, cur ^= 1) {
        wait_ds0();   // WAR: prior fragment ds_loads done before buffer overwrite
        stage_chunk(ga, gb, lane, kc + KCHUNK,
                    &ldsA[cur ^ 1][wave][0], &ldsB[cur ^ 1][wave][0]);
        // async loads retire in order: <=ASYNC_PER_STAGE outstanding means the
        // current chunk's 8 ops (issued earlier) have all landed in LDS
        wait_async<ASYNC_PER_STAGE>();
        asm volatile("" ::: "memory");   // fence compiler across stage/consume boundary
        consume_chunk(&ldsA[cur][wave][r * ROWPAD + hi],
                      &ldsB[cur][wave][r * ROWPAD + hi], acc);
        asm volatile("" ::: "memory");
    }
    // peeled final chunk
    wait_async<0>();
    asm volatile("" ::: "memory");
    consume_chunk(&ldsA[cur][wave][r * ROWPAD + hi],
                  &ldsB[cur][wave][r * ROWPAD + hi], acc);

    // Diagonal of C (ISA 16x16 f32 layout): lanes 0-7 -> accumulator element lane holds diagonal entry lane,
    // lanes 24-31 -> accumulator element lane-24 holds diagonal entry lane-16
    const int sel = (lane < 8) ? lane : ((lane >= 24) ? (lane - 24) : -1);
    const int pp  = (lane < 8) ? lane : ((lane >= 24) ? (lane - 16) : -1);
    float d = 0.0f;
#pragma unroll
    for (int i = 0; i < 8; ++i) d += (sel == i) ? acc[i] : 0.0f;

    const int   p    = (pp >= 0) ? pp : 0;
    const float invA = __shfl(invOwn, p, 32);        // a inv-norm lives at lane p
    const float invB = __shfl(invOwn, 16 + p, 32);   // b inv-norm lives at lane 16+p

    float term = 0.0f;
    if (pp >= 0) {
        const float cosv = d * invA * invB;
        term = fabsf(1.0f - cosv - margin);
    }
#pragma unroll
    for (int off = 16; off > 0; off >>= 1) term += __shfl_xor(term, off, 32);
    if (lane == 0)
        atomicAdd(&slots[(tile & (NSLOT - 1)) * SLOT_STRIDE], term);
}

__global__ void finalize_kernel(const float* __restrict__ slots,
                                float* __restrict__ out) {
    const int lane = threadIdx.x & 31;
    float s = 0.0f;
    for (int i = lane; i < NSLOT; i += 32) s += slots[i * SLOT_STRIDE];
#pragma unroll
    for (int off = 16; off > 0; off >>= 1) s += __shfl_xor(s, off, 32);
    if (lane == 0) out[0] = s / (float)(N_PC + N_SIB);
}

extern "C" void kernel_launch(void* const* d_in, const int* in_sizes, int n_in,
                              void* d_out, int out_size, void* d_ws, size_t ws_size,
                              hipStream_t stream) {
    (void)in_sizes; (void)n_in; (void)out_size; (void)ws_size;
    const float* emb = (const float*)d_in[0];
    const int*   pc  = (const int*)d_in[1];
    const int*   sib = (const int*)d_in[2];
    float* inv_norm = (float*)d_ws;
    float* slots    = inv_norm + NUM_CLASSES;
    float* out      = (float*)d_out;

    zero_kernel<<<(NSLOT * SLOT_STRIDE + 255) / 256, 256, 0, stream>>>(slots);
    inv_norm_kernel<<<(NUM_CLASSES + 7) / 8, 256, 0, stream>>>(emb, inv_norm);
    pair_loss_kernel<<<NT_TOTAL / WAVES_PER_BLOCK, WAVES_PER_BLOCK * 32, 0, stream>>>(
        emb, pc, sib, inv_norm, slots);
    finalize_kernel<<<1, 32, 0, stream>>>(slots, out);
}